// LengthRegulator_44942537785906
// MI455X (gfx1250) — compile-verified
//
#include <hip/hip_runtime.h>
#include <hip/hip_bf16.h>
#include <stdint.h>

// LengthRegulator: B=32, T=512, D=384, DUR_MAX=8, MAX_OUT=T*(DUR_MAX-1)=3584, ALPHA=1.0
// Pure memory-bound gather/expand. ~176MB NT writes + ~25MB reads => ~9us floor @ 23.3 TB/s.

#define LR_B       32
#define LR_T       512
#define LR_D       384
#define LR_MAXOUT  3584
#define WAVES_PER_BLOCK 8   // 256 threads, wave32

#if defined(__HIP_DEVICE_COMPILE__) && defined(__gfx1250__) && \
    __has_builtin(__builtin_amdgcn_global_load_async_to_lds_b128) && \
    __has_builtin(__builtin_amdgcn_s_wait_asynccnt)
#define USE_ASYNC_LDS 1
#else
#define USE_ASYNC_LDS 0
#endif

typedef float lr_v4f __attribute__((ext_vector_type(4)));   // native vector -> b128 + NT hint

#if USE_ASYNC_LDS
typedef int lr_v4i __attribute__((__vector_size__(16)));
typedef __attribute__((address_space(1))) lr_v4i lr_gv4i;   // global
typedef __attribute__((address_space(3))) lr_v4i lr_lv4i;   // LDS
#endif

// ---------------------------------------------------------------------------
// Kernel 1: per-batch inclusive scan of clamped durations.
// patched = max(round(1.0 * d), 0) == d here, but keep the clamp for fidelity.
// One block (512 threads = 16 waves) per batch; Hillis-Steele scan in LDS.
// ---------------------------------------------------------------------------
__global__ __launch_bounds__(LR_T) void lr_scan_kernel(const int* __restrict__ dur,
                                                       int* __restrict__ cum) {
    __shared__ int s[LR_T];
    const int b = blockIdx.x;
    const int t = threadIdx.x;

    int d = dur[b * LR_T + t];
    d = d > 0 ? d : 0;
    s[t] = d;
    __syncthreads();

#pragma unroll
    for (int off = 1; off < LR_T; off <<= 1) {
        int v = (t >= off) ? s[t - off] : 0;
        __syncthreads();
        s[t] += v;
        __syncthreads();
    }
    cum[b * LR_T + t] = s[t];
}

// ---------------------------------------------------------------------------
// Kernel 2: gather/expand. Block = 256 threads (8 wave32s); each wave owns one
// output row. Cum table (2KB) staged in LDS via async-to-LDS (gfx1250 path).
// Branch-free 9-step binary search (searchsorted right), then 3x b128/lane
// row copy: 32 lanes x 16B x 3 = 1536B = D*4 bytes, fully coalesced.
// Output stores are non-temporal: 176MB write-once stream vs 192MB L2 --
// keep the re-read input rows resident instead.
// ---------------------------------------------------------------------------
__global__ __launch_bounds__(256) void lr_gather_kernel(const float* __restrict__ x,
                                                        const int* __restrict__ cum,
                                                        float* __restrict__ out) {
    __shared__ int scum[LR_T];

    const int b    = blockIdx.y;
    const int tid  = threadIdx.x;
    const int wave = tid >> 5;
    const int lane = tid & 31;

#if USE_ASYNC_LDS
    if (wave == 0) {
        lr_gv4i* g = (lr_gv4i*)(cum + b * LR_T);   // global src, v4i units
        lr_lv4i* l = (lr_lv4i*)scum;               // LDS dst, v4i units
#pragma unroll
        for (int k = 0; k < 4; ++k) {
            // 32 lanes x 16B = 512B per op; 4 ops = 2KB cum table.
            __builtin_amdgcn_global_load_async_to_lds_b128(
                g + lane + k * 32, l + lane + k * 32, 0, 0);
        }
        __builtin_amdgcn_s_wait_asynccnt(0);
    }
    __syncthreads();
#else
#pragma unroll
    for (int i = tid; i < LR_T; i += 256) scum[i] = cum[b * LR_T + i];
    __syncthreads();
#endif

    const int total = scum[LR_T - 1];
    const int pos   = blockIdx.x * WAVES_PER_BLOCK + wave;   // grid.x*8 == MAX_OUT

    // searchsorted(cum, pos, side='right'): first idx with scum[idx] > pos.
    int lo = 0, hi = LR_T;
#pragma unroll
    for (int step = 0; step < 9; ++step) {   // 2^9 = 512 = T
        int mid = (lo + hi) >> 1;
        if (scum[mid] > pos) hi = mid; else lo = mid + 1;
    }
    int idx = lo < (LR_T - 1) ? lo : (LR_T - 1);   // clip to T-1
    const bool valid = pos < total;

    const lr_v4f* __restrict__ src =
        (const lr_v4f*)(x + ((size_t)b * LR_T + (size_t)idx) * LR_D);
    lr_v4f* __restrict__ dst =
        (lr_v4f*)(out + ((size_t)b * LR_MAXOUT + (size_t)pos) * LR_D);

    // D = 384 floats = 96 v4f = 32 lanes x 3.
#pragma unroll
    for (int k = 0; k < 3; ++k) {
        lr_v4f v = valid ? src[lane + 32 * k] : (lr_v4f)(0.0f);
        __builtin_nontemporal_store(v, dst + lane + 32 * k);
    }
}

extern "C" void kernel_launch(void* const* d_in, const int* in_sizes, int n_in,
                              void* d_out, int out_size, void* d_ws, size_t ws_size,
                              hipStream_t stream) {
    (void)in_sizes; (void)n_in; (void)out_size; (void)ws_size;

    const float* x   = (const float*)d_in[0];   // (B, T, D) float32
    const int*   dur = (const int*)d_in[1];     // (B, T) int32
    float*       out = (float*)d_out;           // (B, MAX_OUT, D) float32
    int*         cum = (int*)d_ws;              // (B, T) int32 scratch (64KB)

    lr_scan_kernel<<<LR_B, LR_T, 0, stream>>>(dur, cum);

    dim3 grid(LR_MAXOUT / WAVES_PER_BLOCK, LR_B);
    lr_gather_kernel<<<grid, 256, 0, stream>>>(x, cum, out);
}